// GCN_Layer_34591666602118
// MI455X (gfx1250) — compile-verified
//
#include <hip/hip_runtime.h>

// ---------------------------------------------------------------------------
// GCN layer: out = scatter_add( (x@W)[src] * dinv[src]*dinv[dst] ) + b
// MI455X (gfx1250, wave32). GEMM uses V_WMMA_F32_16X16X4_F32 (fp32-exact).
// ---------------------------------------------------------------------------

typedef __attribute__((ext_vector_type(2))) float v2f;
typedef __attribute__((ext_vector_type(8))) float v8f;

#define GCN_D 128

// ---- degree with self loop ------------------------------------------------
__global__ void deg_init_kernel(float* __restrict__ deg, int n) {
  int i = blockIdx.x * blockDim.x + threadIdx.x;
  if (i < n) deg[i] = 1.0f;  // self-loop contributes 1 at every node
}

__global__ void deg_acc_kernel(const int* __restrict__ dst,
                               float* __restrict__ deg, int e) {
  int i = blockIdx.x * blockDim.x + threadIdx.x;
  if (i < e) atomicAdd(&deg[dst[i]], 1.0f);
}

__global__ void dinv_kernel(const float* __restrict__ deg,
                            float* __restrict__ dinv, int n) {
  int i = blockIdx.x * blockDim.x + threadIdx.x;
  if (i < n) {
    float d = deg[i];
    dinv[i] = (d > 0.0f) ? rsqrtf(d) : 0.0f;
  }
}

// ---- h = x @ W via fp32 WMMA ----------------------------------------------
// One wave per 16-row strip; computes all 8 column tiles (full D=128 width)
// so x is streamed from HBM exactly once. K-loop: 32 steps of K=4.
//
// A (16x4 f32, 2 VGPR/lane): lane l, half=l>>4, m=l&15:
//   a.x = x[m][k0+2*half], a.y = x[m][k0+2*half+1]
// B (4x16 f32, 2 VGPR/lane): row = vgpr + 2*half, col n = l&15:
//   b.x = W[k0+2*half][n], b.y = W[k0+2*half+1][n]
// C/D (16x16 f32, 8 VGPR): element v of lane l is h[m0 + v + 8*half][n]
__global__ __launch_bounds__(256) void gemm_xw_wmma(
    const float* __restrict__ x, const float* __restrict__ W,
    float* __restrict__ h, int ntiles) {
  const int lane = threadIdx.x & 31;
  const int wave = threadIdx.x >> 5;
  const int tile = blockIdx.x * (blockDim.x >> 5) + wave;  // wave-uniform
  if (tile >= ntiles) return;                               // whole-wave exit

  const int m0   = tile * 16;
  const int half = lane >> 4;   // 0 or 1
  const int ml   = lane & 15;   // m (for A) / n-within-tile (for B, C/D)

  v8f acc[8] = {};  // 8 column tiles of 16x16 accumulators (64 VGPRs)

  const float* xrow = x + (size_t)(m0 + ml) * GCN_D;

  for (int k0 = 0; k0 < GCN_D; k0 += 4) {
    const int ka = k0 + 2 * half;
    v2f a;
    a.x = xrow[ka];
    a.y = xrow[ka + 1];
    const float* w0 = W + (size_t)ka * GCN_D;
#pragma unroll
    for (int nt = 0; nt < 8; ++nt) {
      const int n = nt * 16 + ml;
      v2f bm;
      bm.x = w0[n];
      bm.y = w0[GCN_D + n];
      acc[nt] = __builtin_amdgcn_wmma_f32_16x16x4_f32(
          /*neg_a=*/false, a, /*neg_b=*/false, bm,
          /*c_mod=*/(short)0, acc[nt], /*reuse_a=*/false, /*reuse_b=*/false);
    }
  }

#pragma unroll
  for (int nt = 0; nt < 8; ++nt) {
    const int n = nt * 16 + ml;
#pragma unroll
    for (int v = 0; v < 8; ++v) {
      h[(size_t)(m0 + v + 8 * half) * GCN_D + n] = acc[nt][v];
    }
  }
}

// Scalar tail for rows not covered by full 16-row WMMA tiles (n % 16 != 0;
// unused for N=100000 but kept for generality).
__global__ void gemm_tail_kernel(const float* __restrict__ x,
                                 const float* __restrict__ W,
                                 float* __restrict__ h, int row0, int n) {
  long idx = (long)blockIdx.x * blockDim.x + threadIdx.x;
  long total = (long)(n - row0) * GCN_D;
  if (idx >= total) return;
  int r = row0 + (int)(idx >> 7);
  int c = (int)(idx & (GCN_D - 1));
  float acc = 0.0f;
  for (int k = 0; k < GCN_D; ++k) acc += x[(size_t)r * GCN_D + k] * W[(size_t)k * GCN_D + c];
  h[(size_t)r * GCN_D + c] = acc;
}

// ---- out = h * dinv^2 (self loop) + b ; also deterministically inits out ---
__global__ void self_bias_kernel(const float* __restrict__ h,
                                 const float* __restrict__ dinv,
                                 const float* __restrict__ b,
                                 float* __restrict__ out, int n) {
  long idx = (long)blockIdx.x * blockDim.x + threadIdx.x;
  if (idx >= (long)n * GCN_D) return;
  int i = (int)(idx >> 7);
  int d = (int)(idx & (GCN_D - 1));
  float di = dinv[i];
  out[idx] = h[idx] * di * di + b[d];
}

// ---- edge scatter: one wave per edge, 4 floats per lane --------------------
// h and out are L2-resident (51MB each << 192MB L2), so the gather + atomic
// RMW traffic stays on-chip.
__global__ __launch_bounds__(256) void scatter_kernel(
    const int* __restrict__ src, const int* __restrict__ dst,
    const float* __restrict__ h, const float* __restrict__ dinv,
    float* __restrict__ out, int e) {
  long t = (long)blockIdx.x * blockDim.x + threadIdx.x;
  int edge = (int)(t >> 5);        // wave-uniform
  if (edge >= e) return;
  int lane = (int)(t & 31);

  int s = src[edge];
  int d = dst[edge];
  float w = dinv[s] * dinv[d];

  const float4 hv = *(const float4*)(h + (size_t)s * GCN_D + lane * 4);
  float* o = out + (size_t)d * GCN_D + lane * 4;
  atomicAdd(o + 0, hv.x * w);
  atomicAdd(o + 1, hv.y * w);
  atomicAdd(o + 2, hv.z * w);
  atomicAdd(o + 3, hv.w * w);
}

// ---------------------------------------------------------------------------
extern "C" void kernel_launch(void* const* d_in, const int* in_sizes, int n_in,
                              void* d_out, int out_size, void* d_ws, size_t ws_size,
                              hipStream_t stream) {
  const float* x    = (const float*)d_in[0];
  const int*   edge = (const int*)d_in[1];
  const float* W    = (const float*)d_in[2];
  const float* b    = (const float*)d_in[3];

  const int n = in_sizes[0] / GCN_D;  // 100000
  const int e = in_sizes[1] / 2;      // 1600000
  const int* src = edge;
  const int* dst = edge + e;

  float* ws   = (float*)d_ws;
  float* h    = ws;                       // N*D floats
  float* deg  = ws + (size_t)n * GCN_D;   // N floats
  float* dinv = deg + n;                  // N floats
  float* out  = (float*)d_out;

  // 1) degrees (self-loop + edges)
  deg_init_kernel<<<(n + 255) / 256, 256, 0, stream>>>(deg, n);
  deg_acc_kernel<<<(e + 255) / 256, 256, 0, stream>>>(dst, deg, e);
  dinv_kernel<<<(n + 255) / 256, 256, 0, stream>>>(deg, dinv, n);

  // 2) h = x @ W   (fp32 WMMA, one wave per 16-row strip)
  const int ntiles = n / 16;
  if (ntiles > 0) {
    const int wavesPerBlock = 8;  // 256 threads
    gemm_xw_wmma<<<(ntiles + wavesPerBlock - 1) / wavesPerBlock, 256, 0, stream>>>(
        x, W, h, ntiles);
  }
  const int row0 = ntiles * 16;
  if (row0 < n) {
    long tail = (long)(n - row0) * GCN_D;
    gemm_tail_kernel<<<(int)((tail + 255) / 256), 256, 0, stream>>>(x, W, h, row0, n);
  }

  // 3) out = h*dinv^2 + b  (initializes every element of out)
  {
    long total = (long)n * GCN_D;
    self_bias_kernel<<<(int)((total + 255) / 256), 256, 0, stream>>>(h, dinv, b, out, n);
  }

  // 4) scatter-add edge messages (one wave per edge)
  {
    long threads = (long)e * 32;
    scatter_kernel<<<(int)((threads + 255) / 256), 256, 0, stream>>>(
        src, dst, h, dinv, out, e);
  }
}